// TPDNEncoder_3367254360395
// MI455X (gfx1250) — compile-verified
//
#include <hip/hip_runtime.h>
#include <hip/hip_bf16.h>

typedef __attribute__((ext_vector_type(2))) float v2f;
typedef __attribute__((ext_vector_type(4))) float v4f;
typedef __attribute__((ext_vector_type(8))) float v8f;

#define BATCH 128
#define SEQ   1024
#define FDIM  256
#define RDIM  64
#define ODIM  1024
#define KDIM  (FDIM * RDIM)   // 16384

static __device__ __forceinline__ v8f wmma_f32(v2f a, v2f b, v8f c) {
    return __builtin_amdgcn_wmma_f32_16x16x4_f32(
        false, a, false, b, (short)0, c, false, false);
}

// ---------------------------------------------------------------------------
// Kernel 1: per-batch masked GEMM  T[b] = (mask ⊙ F[b])^T @ R[b]  (256 x 64)
//   grid = (BATCH, FDIM/64), block = 256 = 8 waves.
//   Block region: E=64 (4 tiles) x R=64 (4 tiles). Wave (we, wrh) computes
//   E-tile we x R-tiles {2*wrh, 2*wrh+1}, sharing one A fragment.
//   LDS is K-pair interleaved: Fp[p][2e + (k&1)] so an A/B fragment is one
//   ds_load_b64. Pair-row stride 160 floats (160 mod 64 == 32 -> the two
//   half-waves, reading pairs p and p+1, hit disjoint bank halves).
// ---------------------------------------------------------------------------
__global__ __launch_bounds__(256) void k1_tp_bind(
    const float* __restrict__ fillers,    // (B, SEQ, FDIM)
    const float* __restrict__ roles,      // (B, SEQ, RDIM)
    const long long* __restrict__ lengths,
    float* __restrict__ T)                // (B, FDIM, RDIM)
{
    constexpr int FS = 160;   // 2*64 + 32 pad (floats per pair-row)
    constexpr int RS = 160;
    __shared__ float Fp[32 * FS];
    __shared__ float Rp[32 * RS];

    const int b    = blockIdx.x;
    const int e0   = blockIdx.y * 64;
    const int tid  = threadIdx.x;
    const int wave = tid >> 5;
    const int lane = tid & 31;
    const int m    = lane & 15;   // M/N index within a 16x16 tile
    const int kh   = lane >> 4;   // half-wave: selects K-pair p or p+1
    const int we   = wave & 3;    // E-tile 0..3
    const int wrh  = wave >> 2;   // R-tile pair 0..1

    const long long len = lengths[b];
    const float* Fb = fillers + (size_t)b * SEQ * FDIM;
    const float* Rb = roles   + (size_t)b * SEQ * RDIM;

    v8f acc0 = {}, acc1 = {};

    for (int s0 = 0; s0 < SEQ; s0 += 64) {
        __syncthreads();
        // stage fillers: 64(k) x 64(e), mask folded in, pair-interleaved.
        // 512 units of (k-pair, 4 e) across 256 threads.
#pragma unroll
        for (int t = 0; t < 2; ++t) {
            int u  = tid + t * 256;      // 0..511
            int p  = u >> 4;             // k-pair 0..31
            int e4 = (u & 15) << 2;      // 0,4,...,60
            int s  = s0 + 2 * p;
            float m0 = ((long long)(s    ) < len) ? 1.0f : 0.0f;
            float m1 = ((long long)(s + 1) < len) ? 1.0f : 0.0f;
            v4f v0 = *(const v4f*)(Fb + (size_t)(s    ) * FDIM + e0 + e4);
            v4f v1 = *(const v4f*)(Fb + (size_t)(s + 1) * FDIM + e0 + e4);
            if (s0 + 64 < SEQ)   // prefetch next chunk of the filler stream
                __builtin_prefetch(Fb + (size_t)(s + 64) * FDIM + e0 + e4, 0, 0);
            v0 *= m0; v1 *= m1;
#pragma unroll
            for (int j = 0; j < 4; ++j) {
                v2f pr = {v0[j], v1[j]};
                *(v2f*)&Fp[p * FS + 2 * (e4 + j)] = pr;
            }
        }
        // stage roles: 64(k) x 64(r), pair-interleaved. 512 units.
#pragma unroll
        for (int t = 0; t < 2; ++t) {
            int u  = tid + t * 256;
            int p  = u >> 4;
            int r4 = (u & 15) << 2;
            int s  = s0 + 2 * p;
            v4f v0 = *(const v4f*)(Rb + (size_t)(s    ) * RDIM + r4);
            v4f v1 = *(const v4f*)(Rb + (size_t)(s + 1) * RDIM + r4);
#pragma unroll
            for (int j = 0; j < 4; ++j) {
                v2f pr = {v0[j], v1[j]};
                *(v2f*)&Rp[p * RS + 2 * (r4 + j)] = pr;
            }
        }
        __syncthreads();

#pragma unroll
        for (int kk = 0; kk < 64; kk += 4) {
            const int p = (kk >> 1) + kh;
            v2f a  = *(const v2f*)&Fp[p * FS + 2 * (we * 16 + m)];
            v2f b0 = *(const v2f*)&Rp[p * RS + 2 * (wrh * 32 + m)];
            v2f b1 = *(const v2f*)&Rp[p * RS + 2 * (wrh * 32 + 16 + m)];
            acc0 = wmma_f32(a, b0, acc0);
            acc1 = wmma_f32(a, b1, acc1);
        }
    }

    // D layout: VGPR i -> M = i + 8*kh; N = lane%16
    const int r0 = wrh * 32 + m;
#pragma unroll
    for (int i = 0; i < 8; ++i) {
        int e = e0 + we * 16 + kh * 8 + i;
        T[((size_t)b * FDIM + e) * RDIM + r0]      = acc0[i];
        T[((size_t)b * FDIM + e) * RDIM + r0 + 16] = acc1[i];
    }
}

// ---------------------------------------------------------------------------
// Kernel 2: partial GEMM  P[slice] += T(128x16384) @ W^T over a 4096-wide
//   K slice. grid = (ODIM/128, BATCH/32, 4 K-slices), block = 256 = 8 waves.
//   Block region: M=32 (2 tiles) x N=128 (8 tiles). Wave (mi, nj) computes
//   M-tile mi x N-tiles {2nj, 2nj+1} sharing one A fragment.
//   Pair-row strides: Tp 96 (mod 64 == 32), Wp 288 (mod 64 == 32).
// ---------------------------------------------------------------------------
__global__ __launch_bounds__(256) void k2_linear(
    const float* __restrict__ T,     // (BATCH, KDIM)
    const float* __restrict__ W,     // (ODIM, KDIM)
    float* __restrict__ P)           // (4, BATCH, ODIM) partials
{
    constexpr int TS = 96;    // 2*32 + 32 pad
    constexpr int WS = 288;   // 2*128 + 32 pad
    __shared__ float Tp[32 * TS];
    __shared__ float Wp[32 * WS];

    const int n0   = blockIdx.x * 128;
    const int m0   = blockIdx.y * 32;
    const int ks   = blockIdx.z * (KDIM / 4);
    const int kEnd = ks + (KDIM / 4);
    const int tid  = threadIdx.x;
    const int wave = tid >> 5;
    const int lane = tid & 31;
    const int m    = lane & 15;
    const int kh   = lane >> 4;
    const int mi   = wave & 1;    // M-tile 0..1
    const int nj   = wave >> 1;   // N-tile pair 0..3

    v8f acc0 = {}, acc1 = {};

    for (int k0 = ks; k0 < kEnd; k0 += 64) {
        __syncthreads();
        // stage T: 32(m) x 64(k) pair-interleaved. 512 v4f (2 per thread).
#pragma unroll
        for (int t = 0; t < 2; ++t) {
            int u  = tid + t * 256;     // 0..511
            int mm = u >> 4;            // 0..31
            int q  = u & 15;            // v4f index along k -> pairs 2q,2q+1
            v4f v = *(const v4f*)(T + (size_t)(m0 + mm) * KDIM + k0 + 4 * q);
            v2f lo = {v.x, v.y}, hi = {v.z, v.w};
            *(v2f*)&Tp[(2 * q    ) * TS + 2 * mm] = lo;
            *(v2f*)&Tp[(2 * q + 1) * TS + 2 * mm] = hi;
        }
        // stage W: 128(n) x 64(k) pair-interleaved. 2048 v4f (8 per thread).
#pragma unroll
        for (int t = 0; t < 8; ++t) {
            int u = tid + t * 256;      // 0..2047
            int n = u >> 4;             // 0..127
            int q = u & 15;
            const float* wp = W + (size_t)(n0 + n) * KDIM + k0 + 4 * q;
            v4f v = *(const v4f*)wp;
            if (k0 + 64 < kEnd)         // prefetch next chunk of W stream
                __builtin_prefetch(wp + 64, 0, 0);
            v2f lo = {v.x, v.y}, hi = {v.z, v.w};
            *(v2f*)&Wp[(2 * q    ) * WS + 2 * n] = lo;
            *(v2f*)&Wp[(2 * q + 1) * WS + 2 * n] = hi;
        }
        __syncthreads();

#pragma unroll
        for (int kk = 0; kk < 64; kk += 4) {
            const int p = (kk >> 1) + kh;
            v2f a  = *(const v2f*)&Tp[p * TS + 2 * (mi * 16 + m)];
            v2f b0 = *(const v2f*)&Wp[p * WS + 2 * (nj * 32 + m)];
            v2f b1 = *(const v2f*)&Wp[p * WS + 2 * (nj * 32 + 16 + m)];
            acc0 = wmma_f32(a, b0, acc0);
            acc1 = wmma_f32(a, b1, acc1);
        }
    }

    float* Ps = P + (size_t)blockIdx.z * BATCH * ODIM;
    const int c0 = n0 + nj * 32 + m;
#pragma unroll
    for (int i = 0; i < 8; ++i) {
        int row = m0 + mi * 16 + kh * 8 + i;
        Ps[(size_t)row * ODIM + c0]      = acc0[i];
        Ps[(size_t)row * ODIM + c0 + 16] = acc1[i];
    }
}

// ---------------------------------------------------------------------------
// Kernel 3: out = P0 + P1 + P2 + P3 + bias  (fixed order -> deterministic)
// ---------------------------------------------------------------------------
__global__ __launch_bounds__(256) void k3_reduce(
    const float* __restrict__ P,
    const float* __restrict__ bias,
    float* __restrict__ out)
{
    const size_t i = (size_t)blockIdx.x * 256 + threadIdx.x;  // < BATCH*ODIM
    const int n = (int)(i & (ODIM - 1));
    const size_t S = (size_t)BATCH * ODIM;
    float s = bias[n];
    s += P[i];
    s += P[i + S];
    s += P[i + 2 * S];
    s += P[i + 3 * S];
    out[i] = s;
}

extern "C" void kernel_launch(void* const* d_in, const int* in_sizes, int n_in,
                              void* d_out, int out_size, void* d_ws, size_t ws_size,
                              hipStream_t stream) {
    (void)in_sizes; (void)n_in; (void)out_size; (void)ws_size;
    const float*     fillers = (const float*)d_in[0];
    const float*     roles   = (const float*)d_in[1];
    const long long* lengths = (const long long*)d_in[2];
    const float*     W       = (const float*)d_in[3];
    const float*     bias    = (const float*)d_in[4];
    float* out = (float*)d_out;

    float* T = (float*)d_ws;                       // 128*256*64 f32 = 8 MiB
    float* P = T + (size_t)BATCH * FDIM * RDIM;    // 4*128*1024 f32 = 2 MiB

    dim3 g1(BATCH, FDIM / 64);            // 128 x 4 = 512 WGs
    k1_tp_bind<<<g1, 256, 0, stream>>>(fillers, roles, lengths, T);

    dim3 g2(ODIM / 128, BATCH / 32, 4);   // 8 x 4 x 4 = 128 WGs
    k2_linear<<<g2, 256, 0, stream>>>(T, W, P);

    dim3 g3((BATCH * ODIM) / 256);        // 512 WGs
    k3_reduce<<<g3, 256, 0, stream>>>(P, bias, out);
}